// ConditionalGMM_24189255811410
// MI455X (gfx1250) — compile-verified
//
#include <hip/hip_runtime.h>
#include <math.h>

typedef __attribute__((ext_vector_type(2))) float v2f;
typedef __attribute__((ext_vector_type(4))) float v4f;
typedef __attribute__((ext_vector_type(8))) float v8f;

#define KC   32
#define DC   2000
#define BC   1024
#define LOG2PI_F 1.8378770664093453f
#define TAU_F    1.0f

// workspace layout (float offsets)
// WS_CC: interleaved coefficients, indexed [d>>1][k][4]:
//   slot0 = c1 (d even), slot1 = c1 (d odd), slot2 = c2 (d even), slot3 = c2 (d odd)
#define WS_CC    0
#define WS_C0K   (2 * DC * KC)
#define WS_LOGPI (2 * DC * KC + KC)
#define WS_S     (2 * DC * KC + 2 * KC)  // (B,K) GEMM result

__device__ __forceinline__ float softplus_f(float v) {
    // numerically stable log(1+exp(v))
    return fmaxf(v, 0.0f) + log1pf(expf(-fabsf(v)));
}

// ---- k0: log_softmax(tau*pi) via wave32 reduction; zero c0k and out[0] ----
__global__ void gmm_init_kernel(const float* __restrict__ pi,
                                float* __restrict__ ws,
                                float* __restrict__ out) {
    int k = threadIdx.x;          // exactly 32 threads = one wave32
    float t = TAU_F * pi[k];
    float m = t;
    for (int off = 16; off > 0; off >>= 1)
        m = fmaxf(m, __shfl_xor(m, off, 32));
    float s = expf(t - m);
    for (int off = 16; off > 0; off >>= 1)
        s += __shfl_xor(s, off, 32);
    ws[WS_LOGPI + k] = t - m - logf(s);
    ws[WS_C0K + k]   = 0.0f;
    if (k == 0) out[0] = 0.0f;
}

// ---- k1: per-(k,d) quadratic coefficients into interleaved (D/2,K,4) layout ----
__global__ void gmm_coeff_kernel(const float* __restrict__ fD,
                                 const float* __restrict__ fmu,
                                 const float* __restrict__ bD,
                                 const float* __restrict__ bmu,
                                 const float* __restrict__ alpha,
                                 float* __restrict__ ws) {
    int i = blockIdx.x * blockDim.x + threadIdx.x;
    if (i >= KC * DC) return;
    int k = i / DC;
    int d = i - k * DC;

    float a  = 1.0f / (1.0f + expf(-10.0f * alpha[i]));
    float fv = softplus_f(fD[i]);
    float bv = softplus_f(bD[d]);
    float fm = fmu[i];
    float bm = bmu[d];

    float rf = a / fv;
    float rb = (1.0f - a) / bv;

    float c2 = -0.5f * (rf + rb);
    float c1 = rf * fm + rb * bm;
    float c0 = -0.5f * (a          * (fm * fm / fv + logf(fv) + LOG2PI_F) +
                        (1.0f - a) * (bm * bm / bv + logf(bv) + LOG2PI_F));

    int base = (d >> 1) * (KC * 4) + k * 4;
    int lo   = d & 1;
    ws[WS_CC + base + lo]     = c1;
    ws[WS_CC + base + 2 + lo] = c2;
    atomicAdd(&ws[WS_C0K + k], c0);
}

// ---- k2: S = X^2 * C2^T + X * C1^T via chained f32 WMMA, one wave per 16x16 tile ----
__global__ void gmm_gemm_kernel(const float* __restrict__ x,
                                float* __restrict__ ws) {
    int tid  = blockIdx.x * blockDim.x + threadIdx.x;
    int wave = tid >> 5;          // 0..127
    int lane = tid & 31;
    int rowTile = wave >> 1;      // 0..63  (batch tiles of 16)
    int colTile = wave & 1;       // 0..1   (component tiles of 16)
    int half = lane >> 4;         // K-dim half select for A/B operands
    int l16  = lane & 15;
    int row  = rowTile * 16 + l16;       // A: lane holds matrix row M
    int n    = colTile * 16 + l16;       // B: lane holds matrix col N

    const float* __restrict__ xr  = x + row * DC;
    // this lane's coefficient base: pair index (dk>>1) advances with d>>1 + half
    const float* __restrict__ ccl = ws + WS_CC + half * (KC * 4) + n * 4;

    v8f acc = {};
#pragma unroll 4
    for (int d = 0; d < DC; d += 4) {
        int dk = d + 2 * half;                            // this lane's K-offset pair
        __builtin_prefetch(xr + d + 64, 0, 1);            // global_prefetch_b8 ahead on x
        v2f ax = *reinterpret_cast<const v2f*>(xr + dk);  // A VGPR0/1 = x[row, dk..dk+1]
        v2f ax2 = ax * ax;
        // one b128: {c1[dk,n], c1[dk+1,n], c2[dk,n], c2[dk+1,n]}
        v4f cc = *reinterpret_cast<const v4f*>(ccl + (d >> 1) * (KC * 4));
        v2f b1; b1.x = cc.x; b1.y = cc.y;
        v2f b2; b2.x = cc.z; b2.y = cc.w;
        // acc += x^2 * c2 ; acc += x * c1   (D = A*B + C, f32 all the way)
        acc = __builtin_amdgcn_wmma_f32_16x16x4_f32(false, ax2, false, b2,
                                                    (short)0, acc, false, false);
        acc = __builtin_amdgcn_wmma_f32_16x16x4_f32(false, ax,  false, b1,
                                                    (short)0, acc, false, false);
    }

    // C/D layout: VGPR r holds M=r (lanes 0-15) / M=r+8 (lanes 16-31), N=lane&15
    float* __restrict__ S = ws + WS_S;
#pragma unroll
    for (int r = 0; r < 8; ++r) {
        int brow = rowTile * 16 + r + 8 * half;
        S[brow * KC + n] = acc[r];
    }
}

// ---- k3: per-row argmax + logsumexp over K=32, mean-NLL via atomicAdd ----
__global__ void gmm_final_kernel(const float* __restrict__ ws,
                                 float* __restrict__ out) {
    int b = blockIdx.x * blockDim.x + threadIdx.x;
    if (b >= BC) return;
    const float* __restrict__ S    = ws + WS_S + b * KC;
    const float* __restrict__ c0k  = ws + WS_C0K;
    const float* __restrict__ lpik = ws + WS_LOGPI;

    float vals[KC];
    float m = -INFINITY;
    int arg = 0;
#pragma unroll
    for (int k = 0; k < KC; ++k) {
        float v = S[k] + c0k[k] + lpik[k];
        vals[k] = v;
        if (v > m) { m = v; arg = k; }   // strict > keeps first max (jnp.argmax)
    }
    float s = 0.0f;
#pragma unroll
    for (int k = 0; k < KC; ++k) s += expf(vals[k] - m);
    float lse = m + logf(s);

    out[1 + b] = (float)arg;
    atomicAdd(out, -lse * (1.0f / (float)BC));
}

extern "C" void kernel_launch(void* const* d_in, const int* in_sizes, int n_in,
                              void* d_out, int out_size, void* d_ws, size_t ws_size,
                              hipStream_t stream) {
    const float* x     = (const float*)d_in[0];  // (B,D)
    const float* fD    = (const float*)d_in[1];  // (K,D)
    const float* fmu   = (const float*)d_in[2];  // (K,D)
    const float* bD    = (const float*)d_in[3];  // (1,D)
    const float* bmu   = (const float*)d_in[4];  // (1,D)
    const float* pi    = (const float*)d_in[5];  // (K,)
    const float* alpha = (const float*)d_in[6];  // (K,D)
    float* out = (float*)d_out;                  // [nll_mean, assignment(0..B-1)]
    float* ws  = (float*)d_ws;

    gmm_init_kernel <<<1, 32, 0, stream>>>(pi, ws, out);
    gmm_coeff_kernel<<<(KC * DC + 255) / 256, 256, 0, stream>>>(fD, fmu, bD, bmu, alpha, ws);
    gmm_gemm_kernel <<<16, 256, 0, stream>>>(x, ws);   // 128 waves = 128 output tiles
    gmm_final_kernel<<<(BC + 255) / 256, 256, 0, stream>>>(ws, out);
}